// Involution2d_266287972633
// MI455X (gfx1250) — compile-verified
//
#include <hip/hip_runtime.h>

// ---------------------------------------------------------------------------
// Involution2d for MI455X (gfx1250, wave32, WMMA).
//   k_dwconv       : depthwise 3x3 (pad 1)                       -> ws.h_dw
//   k_pw_wmma      : 1x1 pointwise 64x64 GEMM, 1 wave = M=64 strip
//                    (4 accumulators share each B load, 4x fewer vmem)
//   k_reduce_wmma  : 64->16 reduce GEMM via v_wmma f32           -> ws.r
//   k_bnstats      : BN(train) mean/var -> scale/bias (deterministic)
//   k_span_invol   : FUSED span GEMM (16->784, v_wmma) + 7x7 involution;
//                    halo staged via async global->LDS (ASYNCcnt path),
//                    avoiding the 205MB kern tensor.
// All f32 to match reference numerics; WMMA shape 16x16x4 F32.
// ---------------------------------------------------------------------------

#define B_   4
#define C_   64
#define H_   128
#define W_   128
#define HW   16384          // H_*W_
#define G_   16
#define KS   7
#define CR_  16
#define KK   49             // KS*KS
#define MK   784            // KK*G_  (== 49 * 16, exactly 49 WMMA M-tiles)
#define EPS_ 1e-5f

typedef __attribute__((ext_vector_type(2))) float v2f;
typedef __attribute__((ext_vector_type(8))) float v8f;

typedef __attribute__((address_space(1))) int g_int;   // global AS
typedef __attribute__((address_space(3))) int l_int;   // LDS AS

#if defined(__gfx1250__) && __has_builtin(__builtin_amdgcn_global_load_async_to_lds_b32)
#define ASYNC_LDS 1
#else
#define ASYNC_LDS 0
#endif

static __device__ __forceinline__ void wait_asynccnt0() {
#if ASYNC_LDS
#if __has_builtin(__builtin_amdgcn_s_wait_asynccnt)
  __builtin_amdgcn_s_wait_asynccnt(0);
#else
  asm volatile("s_wait_asynccnt 0x0" ::: "memory");
#endif
#endif
}

// ---------------------------------------------------------------- depthwise
__global__ __launch_bounds__(256) void k_dwconv(const float* __restrict__ x,
                                                const float* __restrict__ wdw,
                                                float* __restrict__ hdw) {
  int idx = blockIdx.x * 256 + threadIdx.x;     // B*C*HW threads
  int p   = idx & (HW - 1);
  int bc  = idx >> 14;
  int c   = bc & (C_ - 1);
  int y   = p >> 7, xw = p & (W_ - 1);
  const float* wp = wdw + c * 9;
  const float* xp = x + bc * HW;
  float s = 0.f;
#pragma unroll
  for (int ky = 0; ky < 3; ky++) {
    int yy = y + ky - 1;
    if (yy < 0 || yy >= H_) continue;
#pragma unroll
    for (int kx = 0; kx < 3; kx++) {
      int xx = xw + kx - 1;
      if (xx < 0 || xx >= W_) continue;
      s += wp[ky * 3 + kx] * xp[yy * W_ + xx];
    }
  }
  hdw[idx] = s;
}

// ------------------------------------------------------- pointwise 1x1 GEMM
// h_pw[b,o,p] = sum_c w_pw[o,c] * h_dw[b,c,p]
// One wave = full M=64 strip x 16-pixel N tile: 4 WMMAs per K-step share one
// B load (B traffic /4 vs one-tile-per-wave).
__global__ __launch_bounds__(256) void k_pw_wmma(const float* __restrict__ hdw,
                                                 const float* __restrict__ wpw,
                                                 float* __restrict__ hpw) {
  int wave  = (blockIdx.x * 256 + threadIdx.x) >> 5;   // 16384 waves
  int lane  = threadIdx.x & 31;
  int b     = wave >> 10;                // 1024 pixel tiles per image
  int p0    = (wave & 1023) << 4;
  int half  = lane >> 4;                 // A/B: K pair selector
  int lr    = lane & 15;                 // A: row, B/D: column
  const float* A0 = wpw + lr * C_;       // row lr of w_pw (mtile adds 16*C_)
  const float* Bb = hdw + b * C_ * HW + p0 + lr;
  __builtin_prefetch(A0, 0, 3);
  __builtin_prefetch(Bb, 0, 1);
  v8f acc0 = {}, acc1 = {}, acc2 = {}, acc3 = {};
  for (int k0 = 0; k0 < C_; k0 += 4) {
    int ka = k0 + half * 2;
    v2f bb, a0, a1, a2, a3;
    bb.x = Bb[ka * HW];
    bb.y = Bb[(ka + 1) * HW];
    a0.x = A0[ka];            a0.y = A0[ka + 1];
    a1.x = A0[16 * C_ + ka];  a1.y = A0[16 * C_ + ka + 1];
    a2.x = A0[32 * C_ + ka];  a2.y = A0[32 * C_ + ka + 1];
    a3.x = A0[48 * C_ + ka];  a3.y = A0[48 * C_ + ka + 1];
    acc0 = __builtin_amdgcn_wmma_f32_16x16x4_f32(false, a0, false, bb,
                                                 (short)0, acc0, false, false);
    acc1 = __builtin_amdgcn_wmma_f32_16x16x4_f32(false, a1, false, bb,
                                                 (short)0, acc1, false, false);
    acc2 = __builtin_amdgcn_wmma_f32_16x16x4_f32(false, a2, false, bb,
                                                 (short)0, acc2, false, false);
    acc3 = __builtin_amdgcn_wmma_f32_16x16x4_f32(false, a3, false, bb,
                                                 (short)0, acc3, false, false);
  }
  float* Ob = hpw + b * C_ * HW + p0 + lr;
#pragma unroll
  for (int v = 0; v < 8; v++) {
    int row = v + half * 8;              // C/D layout: M = v + 8*half
    Ob[(row)      * HW] = acc0[v];
    Ob[(row + 16) * HW] = acc1[v];
    Ob[(row + 32) * HW] = acc2[v];
    Ob[(row + 48) * HW] = acc3[v];
  }
}

// ------------------------------------------------------------- reduce GEMM
// r[b,cr,p] = sum_c w_reduce[cr,c] * x[b,c,p]   (M = CR = 16, single M tile)
__global__ __launch_bounds__(256) void k_reduce_wmma(const float* __restrict__ x,
                                                     const float* __restrict__ wred,
                                                     float* __restrict__ r) {
  int wave  = (blockIdx.x * 256 + threadIdx.x) >> 5;   // 16384 waves
  int lane  = threadIdx.x & 31;
  int b     = wave >> 10;
  int p0    = (wave & 1023) << 4;
  int half  = lane >> 4;
  int lr    = lane & 15;
  const float* Ab = wred + lr * C_;
  const float* Bb = x + b * C_ * HW + p0 + lr;
  __builtin_prefetch(Ab, 0, 3);
  v8f acc = {};
  for (int k0 = 0; k0 < C_; k0 += 4) {
    int ka = k0 + half * 2;
    v2f a, bb;
    a.x  = Ab[ka];
    a.y  = Ab[ka + 1];
    bb.x = Bb[ka * HW];
    bb.y = Bb[(ka + 1) * HW];
    acc = __builtin_amdgcn_wmma_f32_16x16x4_f32(false, a, false, bb,
                                                (short)0, acc, false, false);
  }
  float* Ob = r + b * CR_ * HW + p0 + lr;
#pragma unroll
  for (int v = 0; v < 8; v++) {
    int row = v + half * 8;
    Ob[row * HW] = acc[v];
  }
}

// -------------------------------------------------- BN training stats (det.)
__global__ __launch_bounds__(256) void k_bnstats(const float* __restrict__ r,
                                                 const float* __restrict__ gamma,
                                                 const float* __restrict__ beta,
                                                 float* __restrict__ stats) {
  __shared__ float ssum[256];
  __shared__ float ssq[256];
  int cr = blockIdx.x;
  float s = 0.f, q = 0.f;
  for (int i = threadIdx.x; i < B_ * HW; i += 256) {
    int b = i >> 14;
    int p = i & (HW - 1);
    float v = r[(b * CR_ + cr) * HW + p];
    s += v;
    q += v * v;
  }
  ssum[threadIdx.x] = s;
  ssq[threadIdx.x]  = q;
  __syncthreads();
  for (int st = 128; st > 0; st >>= 1) {
    if (threadIdx.x < st) {
      ssum[threadIdx.x] += ssum[threadIdx.x + st];
      ssq[threadIdx.x]  += ssq[threadIdx.x + st];
    }
    __syncthreads();
  }
  if (threadIdx.x == 0) {
    float n    = (float)(B_ * HW);
    float mean = ssum[0] / n;
    float var  = ssq[0] / n - mean * mean;   // biased var (torch BN train)
    float sc   = gamma[cr] * rsqrtf(var + EPS_);
    stats[cr]      = sc;
    stats[16 + cr] = beta[cr] - mean * sc;
  }
}

// --------------------------------------- fused span GEMM + 7x7 involution
// Block = one 4x4 pixel tile of one image. 8 waves; each wave owns 2 groups:
// 4 span-GEMM WMMA M-tiles into a private LDS chunk, then the 49-tap
// involution for those groups (same-wave LDS ops are in-order -> no barrier).
__global__ __launch_bounds__(256) void k_span_invol(const float* __restrict__ hpw,
                                                    const float* __restrict__ r,
                                                    const float* __restrict__ stats,
                                                    const float* __restrict__ wspan,
                                                    float* __restrict__ out) {
  __shared__ float s_hpw[C_][100];      // 10x10 halo per channel  (25.6 KB)
  __shared__ float s_rbn[CR_][16];      // BN+ReLU'd reduce tile   ( 1.0 KB)
  __shared__ float s_kern[8][64][16];   // per-wave kern chunk     (32.0 KB)

  int tid = threadIdx.x;
  int blk = blockIdx.x;                 // B * 32 * 32 blocks
  int b   = blk >> 10;
  int ty  = (blk >> 5) & 31;
  int tx  = blk & 31;
  int y0  = ty * 4, x0 = tx * 4;
  const float* hb = hpw + b * C_ * HW;

  // halo stage (zero-padded 7x7 unfold support); async global->LDS path
  for (int i = tid; i < C_ * 100; i += 256) {
    int c  = i / 100;
    int sp = i - c * 100;
    int sy = sp / 10, sx = sp - sy * 10;
    int yy = y0 + sy - 3, xx = x0 + sx - 3;
    bool inb = (yy >= 0 && yy < H_ && xx >= 0 && xx < W_);
#if ASYNC_LDS
    if (inb) {
      __builtin_amdgcn_global_load_async_to_lds_b32(
          (g_int*)(hb + c * HW + yy * W_ + xx),
          (l_int*)&s_hpw[c][sp], 0, 0);
    } else {
      s_hpw[c][sp] = 0.f;
    }
#else
    s_hpw[c][sp] = inb ? hb[c * HW + yy * W_ + xx] : 0.f;
#endif
  }
  // BN + ReLU tile: B operand of the span GEMM (K=CR=16, N=16 pixels)
  {
    int cr = tid >> 4, n = tid & 15;
    int py = n >> 2, px = n & 3;
    float v = r[(b * CR_ + cr) * HW + (y0 + py) * W_ + (x0 + px)];
    v = v * stats[cr] + stats[16 + cr];
    s_rbn[cr][n] = fmaxf(v, 0.f);
  }
  wait_asynccnt0();                     // async halo landed in LDS
  __syncthreads();

  int wv   = tid >> 5;
  int lane = tid & 31;
  int half = lane >> 4;
  int lr   = lane & 15;

  for (int gi = 0; gi < 2; gi++) {
    int g = wv * 2 + gi;
    // span GEMM: rows g*49 .. g*49+63 (rows >= 49 in the chunk are scratch)
    for (int t = 0; t < 4; t++) {
      int row = g * 49 + t * 16 + lr;
      row = (row > MK - 1) ? (MK - 1) : row;   // clamp (clamped rows unread)
      const float* Ab = wspan + row * CR_;
      __builtin_prefetch(Ab, 0, 1);
      v8f acc = {};
#pragma unroll
      for (int ks = 0; ks < 4; ks++) {
        int ka = ks * 4 + half * 2;
        v2f a, bb;
        a.x  = Ab[ka];
        a.y  = Ab[ka + 1];
        bb.x = s_rbn[ka][lr];
        bb.y = s_rbn[ka + 1][lr];
        acc = __builtin_amdgcn_wmma_f32_16x16x4_f32(false, a, false, bb,
                                                    (short)0, acc, false, false);
      }
#pragma unroll
      for (int v = 0; v < 8; v++)
        s_kern[wv][t * 16 + v + half * 8][lr] = acc[v];
    }
    // involution: lane -> pixel n = lr, channels cg = half*2 + {0,1}
    int n  = lr;
    int py = n >> 2, px = n & 3;
#pragma unroll
    for (int j = 0; j < 2; j++) {
      int cg = half * 2 + j;
      int c  = g * 4 + cg;
      float s = 0.f;
#pragma unroll
      for (int kk = 0; kk < KK; kk++) {
        int dy = kk / 7, dx = kk - dy * 7;
        s += s_kern[wv][kk][n] * s_hpw[c][(py + dy) * 10 + (px + dx)];
      }
      out[((b * C_ + c) * H_ + (y0 + py)) * W_ + (x0 + px)] = s;
    }
  }
}

// ---------------------------------------------------------------------------
extern "C" void kernel_launch(void* const* d_in, const int* in_sizes, int n_in,
                              void* d_out, int out_size, void* d_ws, size_t ws_size,
                              hipStream_t stream) {
  const float* x     = (const float*)d_in[0];   // input       [B,C,H,W]
  const float* wdw   = (const float*)d_in[1];   // w_dw        [C,1,3,3]
  const float* wpw   = (const float*)d_in[2];   // w_pw        [C,C]
  const float* gamma = (const float*)d_in[3];   // bn_gamma    [CR]
  const float* beta  = (const float*)d_in[4];   // bn_beta     [CR]
  const float* wred  = (const float*)d_in[5];   // w_reduce    [CR,C]
  const float* wspan = (const float*)d_in[6];   // w_span      [K*K*G,CR]
  float* out = (float*)d_out;
  float* ws  = (float*)d_ws;

  float* hdw   = ws;                       // B*C*HW   = 4194304 floats
  float* hpw   = ws + 4194304;             // B*C*HW
  float* rr    = ws + 8388608;             // B*CR*HW  = 1048576 floats
  float* stats = ws + 9437184;             // scale[16], bias[16]

  k_dwconv     <<<(B_ * C_ * HW) / 256, 256, 0, stream>>>(x, wdw, hdw);
  k_pw_wmma    <<<2048, 256, 0, stream>>>(hdw, wpw, hpw);
  k_reduce_wmma<<<2048, 256, 0, stream>>>(x, wred, rr);
  k_bnstats    <<<CR_, 256, 0, stream>>>(rr, gamma, beta, stats);
  k_span_invol <<<B_ * 32 * 32, 256, 0, stream>>>(hpw, rr, stats, wspan, out);
}